// CosineBlock_27324581937367
// MI455X (gfx1250) — compile-verified
//
#include <hip/hip_runtime.h>
#include <hip/hip_bf16.h>
#include <math.h>

// CDNA5 / gfx1250: wave32, WMMA F32 16x16x4 (fp32 in/out -> bit-compatible
// precision with the fp32 reference). Memory-bound kernel (~577 MB @ 23.3 TB/s
// => ~25 us floor); WMMA does the batched per-pixel dot products via the
// diagonal of A(x2 tile) x B(x1 tile), sums-of-squares ride along on VALU.

typedef __attribute__((ext_vector_type(2))) float v2f;
typedef __attribute__((ext_vector_type(8))) float v8f;

// Only check in the DEVICE pass: amdgcn builtins are invisible to the host pass.
#if defined(__gfx1250__)
#if !__has_builtin(__builtin_amdgcn_wmma_f32_16x16x4_f32)
#error "missing __builtin_amdgcn_wmma_f32_16x16x4_f32 on gfx1250 device pass"
#endif
#endif

#define EPS 1e-12f
#define BATCH 32
#define SDIM 8
#define CDIM 512
#define PIX 1024          // 32*32 pixels per (b,c) plane, contiguous
#define TILES_PER_B 64    // 1024 / 16
#define CSPLIT 4          // waves per block; each wave covers CDIM/CSPLIT channels
#define CPW (CDIM / CSPLIT)   // 128
#define CSTEPS (CPW / 4)      // 32 K=4 WMMA steps per wave

__global__ __launch_bounds__(128)
void cosdist_wmma_kernel(const float* __restrict__ x1,
                         const float* __restrict__ x2,
                         float* __restrict__ out)
{
  const int tid  = threadIdx.x;
  const int lane = tid & 31;
  const int wave = tid >> 5;          // 0..3, channel split
  const int bidx = blockIdx.x;        // 0..2047
  const int b    = bidx >> 6;
  const int p0   = (bidx & 63) << 4;  // 16-pixel tile base

  // WMMA f32 16x16x4 operand layout (ISA 7.12.2):
  //   A (16x4, MxK): lane l -> M = l&15 ; VGPR0 holds K = (l<16?0:2), VGPR1 = K+1
  //   B (4x16, KxN): lane l -> N = l&15 ; VGPR0 holds K = (l<16?0:2), VGPR1 = K+1
  // => identical per-lane addressing for both operands: pixel = p0+(l&15),
  //    channel = cbase + 4*cs + khalf (+1 for second VGPR).
  const int mm    = lane & 15;
  const int khalf = (lane >> 4) << 1;   // 0 or 2
  const int cbase = wave * CPW;

  const float* qptr = x1 + ((size_t)b * CDIM + cbase + khalf) * PIX + p0 + mm;
  const float* aptr = x2 + (((size_t)b * SDIM) * CDIM + cbase + khalf) * PIX + p0 + mm;

  v8f acc[SDIM];
  float ssq[SDIM];
  const v8f vzero = {0.f, 0.f, 0.f, 0.f, 0.f, 0.f, 0.f, 0.f};
#pragma unroll
  for (int s = 0; s < SDIM; ++s) { acc[s] = vzero; ssq[s] = 0.f; }
  float qss = 0.f;

  for (int cs = 0; cs < CSTEPS; ++cs) {
    const float* qp = qptr + (size_t)cs * 4 * PIX;
    const float b0 = qp[0];         // channel c0+khalf     (coalesced 64B rows)
    const float b1 = qp[PIX];       // channel c0+khalf+1
    qss = fmaf(b0, b0, qss);
    qss = fmaf(b1, b1, qss);
    v2f Bv; Bv.x = b0; Bv.y = b1;
#pragma unroll
    for (int s = 0; s < SDIM; ++s) {
      const float* ap = aptr + (size_t)s * (CDIM * PIX) + (size_t)cs * 4 * PIX;
      const float a0 = ap[0];
      const float a1 = ap[PIX];
      ssq[s] = fmaf(a0, a0, ssq[s]);
      ssq[s] = fmaf(a1, a1, ssq[s]);
      v2f Av; Av.x = a0; Av.y = a1;
      // D[m][n] += sum_k x2[s, c0+k, p0+m] * x1[c0+k, p0+n]; diagonal = dots
      acc[s] = __builtin_amdgcn_wmma_f32_16x16x4_f32(
          false, Av, false, Bv, (short)0, acc[s], false, false);
    }
  }

  // Diagonal extraction from C/D layout: diag m lives at
  //   m<8 : VGPR m,   lane m
  //   m>=8: VGPR m-8, lane m+16
  // Lane m (0..15) gathers register (m&7) from srcl.
  const int srcl = (lane < 8) ? lane : (lane + 16);
  float dots[SDIM];
#pragma unroll
  for (int s = 0; s < SDIM; ++s) {
    float d = 0.f;
#pragma unroll
    for (int r = 0; r < 8; ++r) {
      float t = __shfl(acc[s][r], srcl, 32);
      d = ((lane & 7) == r) ? t : d;
    }
    dots[s] = d;
  }

  // Per-pixel sums of squares: halves live in lane m and lane m+16.
  const float qq = qss + __shfl_xor(qss, 16, 32);
  float sspix[SDIM];
#pragma unroll
  for (int s = 0; s < SDIM; ++s)
    sspix[s] = ssq[s] + __shfl_xor(ssq[s], 16, 32);

  // Cross-wave (channel-split) combine via LDS.
  __shared__ float s_dot[CSPLIT][SDIM][16];
  __shared__ float s_ss [CSPLIT][SDIM][16];
  __shared__ float s_qq [CSPLIT][16];
  if (lane < 16) {
#pragma unroll
    for (int s = 0; s < SDIM; ++s) {
      s_dot[wave][s][lane] = dots[s];
      s_ss [wave][s][lane] = sspix[s];
    }
    s_qq[wave][lane] = qq;
  }
  __syncthreads();

  // 128 threads -> 8 s * 16 pixels outputs for this tile.
  const int so = tid >> 4;
  const int mo = tid & 15;
  float dsum = 0.f, ssum = 0.f, qsum = 0.f;
#pragma unroll
  for (int w = 0; w < CSPLIT; ++w) {
    dsum += s_dot[w][so][mo];
    ssum += s_ss [w][so][mo];
    qsum += s_qq [w][mo];
  }
  const float denom = fmaxf(sqrtf(qsum), EPS) * fmaxf(sqrtf(ssum), EPS);
  out[(size_t)b * (SDIM * PIX) + (size_t)so * PIX + p0 + mo] = 1.0f - dsum / denom;
}

extern "C" void kernel_launch(void* const* d_in, const int* in_sizes, int n_in,
                              void* d_out, int out_size, void* d_ws, size_t ws_size,
                              hipStream_t stream) {
  (void)in_sizes; (void)n_in; (void)out_size; (void)d_ws; (void)ws_size;
  const float* x1 = (const float*)d_in[0];   // [32, 512, 32, 32] fp32
  const float* x2 = (const float*)d_in[1];   // [32, 8, 512, 32, 32] fp32
  float* out = (float*)d_out;                // [32, 8192] fp32
  dim3 grid(BATCH * TILES_PER_B);            // 2048 blocks: one 16-pixel tile each
  dim3 block(128);                           // 4 waves, channel-split by 128
  hipLaunchKernelGGL(cosdist_wmma_kernel, grid, block, 0, stream, x1, x2, out);
}